// GAE_47991964565537
// MI455X (gfx1250) — compile-verified
//
#include <hip/hip_runtime.h>
#include <hip/hip_bf16.h>

typedef __attribute__((ext_vector_type(2))) float v2f;
typedef __attribute__((ext_vector_type(8))) float v8f;

#define NUM_USER 1000
#define NUM_NODES 3000
#define NUM_REL 5
#define HID 500
#define OUT 75
#define OUTP 80          // OUT padded to multiple of 16
#define NUM_EDGES 200000
#define NUM_ITEM 2000
#define MPADU 1008       // NUM_USER padded to multiple of 16

// ---- ws layout (in floats) ----
#define OFF_WEIGHT 0ul                       // R*N*H           = 7,500,000
#define OFF_AGG    7500000ul                 // N*H             = 1,500,000
#define OFF_Z      9000000ul                 // 3000*OUTP       =   240,000
#define OFF_QT     9240000ul                 // R*OUTP*OUTP     =    32,000
#define OFF_WT     9272000ul                 // OUTP*HID        =    40,000
#define OFF_AALL   9312000ul                 // R*MPADU*OUTP    =   403,200
// total: 9,715,200 floats = 38.9 MB

static __device__ __forceinline__ v2f ldv2(const float* p) {
    float2 t = *(const float2*)p;
    v2f r; r.x = t.x; r.y = t.y; return r;
}

// K1: ordinal cumulative weight sharing (vectorized): weight[r] = cumsum ord_basis
__global__ void cumsum_kernel(const float4* __restrict__ ob, float4* __restrict__ w) {
    const long nh4 = (long)NUM_NODES * HID / 4;
    long t = (long)blockIdx.x * blockDim.x + threadIdx.x;
    if (t >= nh4) return;
    float4 s = make_float4(0.f, 0.f, 0.f, 0.f);
    #pragma unroll
    for (int r = 0; r < NUM_REL; ++r) {
        float4 v = ob[(long)r * nh4 + t];
        s.x += v.x; s.y += v.y; s.z += v.z; s.w += v.w;
        w[(long)r * nh4 + t] = s;
    }
}

// K2: per-edge gather * norm, scatter-add into agg (L2-resident float atomics)
__global__ void scatter_kernel(const float* __restrict__ w,
                               const int* __restrict__ ei,
                               const int* __restrict__ et,
                               const float* __restrict__ norm,
                               const int* __restrict__ x,
                               float* __restrict__ agg) {
    int e = blockIdx.x;
    int src = ei[e];
    int dst = ei[NUM_EDGES + e];
    int t = et[e];
    float nv = norm[e];
    const float* wrow = w + ((long)t * NUM_NODES + x[src]) * HID;
    float* arow = agg + (long)dst * HID;
    for (int h = threadIdx.x; h < HID; h += blockDim.x)
        atomicAdd(&arow[h], wrow[h] * nv);
}

// K3a: W transposed + padded: WT[n][k] = W[k][n], zero for n>=OUT
__global__ void wt_kernel(const float* __restrict__ W, float* __restrict__ wt) {
    int t = blockIdx.x * blockDim.x + threadIdx.x;
    if (t >= OUTP * HID) return;
    int n = t / HID, k = t % HID;
    wt[t] = (n < OUT) ? W[(long)k * OUT + n] : 0.f;
}

// K3b: dense layer: z = relu(relu(agg) @ W), (3000 x OUTP), pad cols -> 0.
// One wave per 16x16 tile; unconditional float2 loads; 4 independent acc chains.
__global__ void __launch_bounds__(32) dense_wmma(const float* __restrict__ agg,
                                                 const float* __restrict__ wt,
                                                 float* __restrict__ z) {
    int l = threadIdx.x;
    int m0 = blockIdx.x * 16;
    int n0 = blockIdx.y * 16;
    int m = l & 15;
    int khalf = (l >> 4);
    int kof = khalf * 2;
    const float* arow = agg + (long)(m0 + m) * HID + kof;  // A rows >=3000 read ws junk; stores guarded
    const float* brow = wt + (long)(n0 + m) * HID + kof;
    v8f c0 = {}, c1 = {}, c2 = {}, c3 = {};
    int k0 = 0;
    for (; k0 + 16 <= HID; k0 += 16) {
        v2f a, b;
        a = ldv2(arow + k0);      a.x = fmaxf(a.x, 0.f); a.y = fmaxf(a.y, 0.f);
        b = ldv2(brow + k0);
        c0 = __builtin_amdgcn_wmma_f32_16x16x4_f32(false, a, false, b, (short)0, c0, false, false);
        a = ldv2(arow + k0 + 4);  a.x = fmaxf(a.x, 0.f); a.y = fmaxf(a.y, 0.f);
        b = ldv2(brow + k0 + 4);
        c1 = __builtin_amdgcn_wmma_f32_16x16x4_f32(false, a, false, b, (short)0, c1, false, false);
        a = ldv2(arow + k0 + 8);  a.x = fmaxf(a.x, 0.f); a.y = fmaxf(a.y, 0.f);
        b = ldv2(brow + k0 + 8);
        c2 = __builtin_amdgcn_wmma_f32_16x16x4_f32(false, a, false, b, (short)0, c2, false, false);
        a = ldv2(arow + k0 + 12); a.x = fmaxf(a.x, 0.f); a.y = fmaxf(a.y, 0.f);
        b = ldv2(brow + k0 + 12);
        c3 = __builtin_amdgcn_wmma_f32_16x16x4_f32(false, a, false, b, (short)0, c3, false, false);
    }
    for (; k0 < HID; k0 += 4) {   // tail: 500 = 31*16 + 4
        v2f a = ldv2(arow + k0);  a.x = fmaxf(a.x, 0.f); a.y = fmaxf(a.y, 0.f);
        v2f b = ldv2(brow + k0);
        c0 = __builtin_amdgcn_wmma_f32_16x16x4_f32(false, a, false, b, (short)0, c0, false, false);
    }
    #pragma unroll
    for (int v = 0; v < 8; ++v) {
        int row = m0 + v + (khalf << 3);
        if (row < NUM_NODES)
            z[(long)row * OUTP + n0 + m] = fmaxf((c0[v] + c1[v]) + (c2[v] + c3[v]), 0.f);
    }
}

// K4: QT_pad[r][e][d] = sum_b coefs[r,b]*basis[b, d*OUT+e]  (transposed, zero pad)
__global__ void qt_kernel(const float* __restrict__ coefs,
                          const float* __restrict__ basis,
                          float* __restrict__ qt) {
    int t = blockIdx.x * blockDim.x + threadIdx.x;
    if (t >= NUM_REL * OUTP * OUTP) return;
    int r = t / (OUTP * OUTP);
    int rem = t % (OUTP * OUTP);
    int e = rem / OUTP, d = rem % OUTP;
    float v = 0.f;
    if (d < OUT && e < OUT) {
        int de = d * OUT + e;
        v = coefs[r * 2 + 0] * basis[de] + coefs[r * 2 + 1] * basis[OUT * OUT + de];
    }
    qt[t] = v;
}

// K5: A_all[r] = u @ Q_r   (u = z rows [0,1000), K = OUTP; 2 acc chains)
__global__ void __launch_bounds__(32) aall_wmma(const float* __restrict__ z,
                                                const float* __restrict__ qt,
                                                float* __restrict__ aall) {
    int l = threadIdx.x;
    int m0 = blockIdx.x * 16;
    int e0 = blockIdx.y * 16;
    int r  = blockIdx.z;
    int m = l & 15;
    int khalf = (l >> 4);
    int kof = khalf * 2;
    const float* arow = z + (long)(m0 + m) * OUTP + kof;                    // rows <=1007 valid
    const float* brow = qt + ((long)r * OUTP + e0 + m) * OUTP + kof;        // QT[r][e][d]
    v8f c0 = {}, c1 = {};
    for (int k0 = 0; k0 < OUTP; k0 += 8) {
        v2f a, b;
        a = ldv2(arow + k0);     b = ldv2(brow + k0);
        c0 = __builtin_amdgcn_wmma_f32_16x16x4_f32(false, a, false, b, (short)0, c0, false, false);
        a = ldv2(arow + k0 + 4); b = ldv2(brow + k0 + 4);
        c1 = __builtin_amdgcn_wmma_f32_16x16x4_f32(false, a, false, b, (short)0, c1, false, false);
    }
    #pragma unroll
    for (int v = 0; v < 8; ++v) {
        int row = m0 + v + (khalf << 3);
        aall[((long)r * MPADU + row) * OUTP + e0 + m] = c0[v] + c1[v];
    }
}

// K6: out[u,i,r] = A_all[r][u,:] . i[i,:]   (i = z rows [1000,3000))
// B fragment reused across the 5 relations -> 5 independent WMMA chains.
__global__ void __launch_bounds__(32) bidec_wmma(const float* __restrict__ aall,
                                                 const float* __restrict__ z,
                                                 float* __restrict__ out) {
    int l = threadIdx.x;
    int u0 = blockIdx.x * 16;
    int i0 = blockIdx.y * 16;
    int m = l & 15;
    int khalf = (l >> 4);
    int kof = khalf * 2;
    const float* brow = z + (long)(NUM_USER + i0 + m) * OUTP + kof;
    const float* arow = aall + (long)(u0 + m) * OUTP + kof;
    v8f c[NUM_REL];
    #pragma unroll
    for (int r = 0; r < NUM_REL; ++r) c[r] = (v8f){};
    for (int k0 = 0; k0 < OUTP; k0 += 4) {
        v2f b = ldv2(brow + k0);
        #pragma unroll
        for (int r = 0; r < NUM_REL; ++r) {
            v2f a = ldv2(arow + (long)r * MPADU * OUTP + k0);
            c[r] = __builtin_amdgcn_wmma_f32_16x16x4_f32(false, a, false, b, (short)0, c[r], false, false);
        }
    }
    int iidx = i0 + m;
    #pragma unroll
    for (int v = 0; v < 8; ++v) {
        int uidx = u0 + v + (khalf << 3);
        if (uidx < NUM_USER) {
            float* p = out + ((long)uidx * NUM_ITEM + iidx) * NUM_REL;
            #pragma unroll
            for (int r = 0; r < NUM_REL; ++r) p[r] = c[r][v];
        }
    }
}

extern "C" void kernel_launch(void* const* d_in, const int* in_sizes, int n_in,
                              void* d_out, int out_size, void* d_ws, size_t ws_size,
                              hipStream_t stream) {
    const int*   x      = (const int*)d_in[0];
    const int*   ei     = (const int*)d_in[1];
    const int*   et     = (const int*)d_in[2];
    const float* enorm  = (const float*)d_in[3];
    const float* ob     = (const float*)d_in[4];
    const float* Wd     = (const float*)d_in[5];
    const float* basis  = (const float*)d_in[6];
    const float* coefs  = (const float*)d_in[7];
    float* out = (float*)d_out;

    float* ws   = (float*)d_ws;
    float* wgt  = ws + OFF_WEIGHT;
    float* agg  = ws + OFF_AGG;
    float* z    = ws + OFF_Z;
    float* qt   = ws + OFF_QT;
    float* wt   = ws + OFF_WT;
    float* aall = ws + OFF_AALL;

    // 1) ordinal cumsum weights (float4 vectorized)
    long nh = (long)NUM_NODES * HID;
    cumsum_kernel<<<dim3((unsigned)((nh / 4 + 255) / 256)), dim3(256), 0, stream>>>(
        (const float4*)ob, (float4*)wgt);

    // 2) zero agg, then edge scatter-add
    hipMemsetAsync(agg, 0, nh * sizeof(float), stream);
    scatter_kernel<<<dim3(NUM_EDGES), dim3(256), 0, stream>>>(wgt, ei, et, enorm, x, agg);

    // 3) pad/transpose W, then dense layer (WMMA f32)
    wt_kernel<<<dim3((OUTP * HID + 255) / 256), dim3(256), 0, stream>>>(Wd, wt);
    dense_wmma<<<dim3((NUM_NODES + 15) / 16, OUTP / 16), dim3(32), 0, stream>>>(agg, wt, z);

    // 4) QT_pad
    qt_kernel<<<dim3((NUM_REL * OUTP * OUTP + 255) / 256), dim3(256), 0, stream>>>(coefs, basis, qt);

    // 5) A_all[r] = u @ Q_r (WMMA f32)
    aall_wmma<<<dim3(MPADU / 16, OUTP / 16, NUM_REL), dim3(32), 0, stream>>>(z, qt, aall);

    // 6) bilinear decoder (WMMA f32), writes (U*I, R)
    bidec_wmma<<<dim3(MPADU / 16, NUM_ITEM / 16), dim3(32), 0, stream>>>(aall, z, out);
    (void)in_sizes; (void)n_in; (void)out_size; (void)ws_size;
}